// T_GConvGRU_18485539242713
// MI455X (gfx1250) — compile-verified
//
#include <hip/hip_runtime.h>
#include <cstdint>

typedef float v2f __attribute__((ext_vector_type(2)));
typedef float v8f __attribute__((ext_vector_type(8)));

#define CIN  16
#define COUT 64
#define TB   256

// ---------------- WMMA helpers (V_WMMA_F32_16X16X4_F32, wave32) ----------------
// D(16x16,f32) = A(16x4,f32) * B(4x16,f32) + C
__device__ __forceinline__ v8f wmma4(v2f a, v2f b, v8f c) {
  return __builtin_amdgcn_wmma_f32_16x16x4_f32(false, a, false, b, (short)0, c,
                                               false, false);
}

// A fragment: row-major activations, 16 rows starting at row0, K-slice [k0,k0+4)
// lane L: row = L&15 ; VGPR j holds K = k0 + 2*(L>>4) + j  -> contiguous float2
__device__ __forceinline__ v2f load_a4(const float* __restrict__ m, int ld,
                                       int row0, int k0, int lane) {
  const int r = lane & 15, half = lane >> 4;
  const float* p = m + (size_t)(row0 + r) * ld + k0 + 2 * half;
  v2f v; v.x = p[0]; v.y = p[1];
  return v;
}

// B fragment: row-major weights [K x ld], K-slice [k0,k0+4), col tile [n0,n0+16)
// lane L: col = n0 + (L&15) ; VGPR j holds K = k0 + 2*(L>>4) + j
__device__ __forceinline__ v2f load_b4(const float* __restrict__ w, int ld,
                                       int k0, int n0, int lane) {
  const int n = lane & 15, half = lane >> 4;
  const float* p = w + (size_t)(k0 + 2 * half) * ld + n0 + n;
  v2f v; v.x = p[0]; v.y = p[ld];
  return v;
}

// ---------------- small utility kernels ----------------
__global__ void fill0_k(float* __restrict__ p, long long n) {
  long long i = (long long)blockIdx.x * blockDim.x + threadIdx.x;
  if (i < n) p[i] = 0.0f;
}

__global__ void negcopy_k(float* __restrict__ o, const float* __restrict__ a,
                          long long n) {
  long long i = (long long)blockIdx.x * blockDim.x + threadIdx.x;
  if (i < n) o[i] = -a[i];
}

// deg[src] += w (self loops removed)
__global__ void deg_scatter_k(const float* __restrict__ ew,
                              const long long* __restrict__ ei,
                              float* __restrict__ deg, int E) {
  int e = blockIdx.x * blockDim.x + threadIdx.x;
  if (e >= E) return;
  long long s = ei[e], d = ei[(size_t)E + e];
  float w = (s == d) ? 0.0f : ew[e];
  if (w != 0.0f) atomicAdd(deg + s, w);
}

// deg -> dinv (in place)
__global__ void dinv_k(float* __restrict__ deg, int n) {
  int i = blockIdx.x * blockDim.x + threadIdx.x;
  if (i >= n) return;
  float v = deg[i];
  deg[i] = (v > 0.0f) ? rsqrtf(v) : 0.0f;
}

// nw[e] = -dinv[src]*w*dinv[dst]
__global__ void nw_k(const float* __restrict__ ew,
                     const long long* __restrict__ ei,
                     const float* __restrict__ dinv, float* __restrict__ nw,
                     int E) {
  int e = blockIdx.x * blockDim.x + threadIdx.x;
  if (e >= E) return;
  long long s = ei[e], d = ei[(size_t)E + e];
  float w = (s == d) ? 0.0f : ew[e];
  nw[e] = -dinv[s] * w * dinv[d];
}

// out[dst,:] += scale*nw[e]*v[src,:]   (4-channel float4 groups per thread)
// gshift = log2(C/4)
__global__ void prop_scatter_k(const float* __restrict__ v,
                               const float* __restrict__ nw,
                               const long long* __restrict__ ei,
                               float* __restrict__ out, float scale, int E,
                               int C, int gshift) {
  unsigned t = blockIdx.x * blockDim.x + threadIdx.x;
  unsigned e = t >> gshift;
  if (e >= (unsigned)E) return;
  unsigned g = (t & ((1u << gshift) - 1u)) << 2;
  float w = nw[e] * scale;
  if (w == 0.0f) return;
  int s = (int)ei[e];
  int d = (int)ei[(size_t)E + e];
  float4 vv = *(const float4*)(v + (size_t)s * C + g);
  float* po = out + (size_t)d * C + g;
  atomicAdd(po + 0, w * vv.x);
  atomicAdd(po + 1, w * vv.y);
  atomicAdd(po + 2, w * vv.z);
  atomicAdd(po + 3, w * vv.w);
}

// ---------------- fused gate kernels (WMMA f32 16x16x4) ----------------
// One wave computes a 16x64 pre-activation tile for one gate:
//   pre = [T0|T1|T2]_x @ Wx[g] + [T0|T1|T2]_h @ Wh[g] + bx[g] + bh[g]
// gate 0 -> z = sigmoid(pre) ; gate 1 -> hr = h0 * sigmoid(pre)
__global__ void gate_zr_k(const float* __restrict__ x,
                          const float* __restrict__ tx1,
                          const float* __restrict__ tx2,
                          const float* __restrict__ h0,
                          const float* __restrict__ th1,
                          const float* __restrict__ th2,
                          const float* __restrict__ Wx,
                          const float* __restrict__ Wh,
                          const float* __restrict__ bx,
                          const float* __restrict__ bh,
                          float* __restrict__ zout, float* __restrict__ hrout,
                          int n) {
  int wv = (int)(((long long)blockIdx.x * blockDim.x + threadIdx.x) >> 5);
  int lane = threadIdx.x & 31;
  int ntiles = n >> 4;  // n is a multiple of 16 (50000)
  int gate = wv / ntiles;
  if (gate >= 2) return;  // wave-uniform
  int row0 = (wv - gate * ntiles) << 4;

  v8f acc[4] = {};
  const float* xs[3] = {x, tx1, tx2};
  const float* hs[3] = {h0, th1, th2};

#pragma unroll
  for (int kc = 0; kc < 3; ++kc) {
    const float* A = xs[kc];
    const float* W = Wx + (size_t)(gate * 3 + kc) * (CIN * COUT);
    for (int k0 = 0; k0 < CIN; k0 += 4) {
      v2f a = load_a4(A, CIN, row0, k0, lane);
#pragma unroll
      for (int nt = 0; nt < 4; ++nt)
        acc[nt] = wmma4(a, load_b4(W, COUT, k0, nt * 16, lane), acc[nt]);
    }
  }
#pragma unroll
  for (int kc = 0; kc < 3; ++kc) {
    const float* A = hs[kc];
    const float* W = Wh + (size_t)(gate * 3 + kc) * (COUT * COUT);
    for (int k0 = 0; k0 < COUT; k0 += 4) {
      v2f a = load_a4(A, COUT, row0, k0, lane);
#pragma unroll
      for (int nt = 0; nt < 4; ++nt)
        acc[nt] = wmma4(a, load_b4(W, COUT, k0, nt * 16, lane), acc[nt]);
    }
  }

  const int half = lane >> 4, nloc = lane & 15;
#pragma unroll
  for (int nt = 0; nt < 4; ++nt) {
    int col = nt * 16 + nloc;
    float bias = bx[gate * COUT + col] + bh[gate * COUT + col];
#pragma unroll
    for (int i = 0; i < 8; ++i) {
      int row = row0 + i + 8 * half;
      float pre = acc[nt][i] + bias;
      float sg = 1.0f / (1.0f + expf(-pre));
      size_t idx = (size_t)row * COUT + col;
      if (gate == 0)
        zout[idx] = sg;
      else
        hrout[idx] = h0[idx] * sg;
    }
  }
}

// gate 2: htil = tanh([Tx]@Wx[2] + [Thr]@Wh[2] + b) ; h = z*h0 + (1-z)*htil
__global__ void gate_h_k(const float* __restrict__ x,
                         const float* __restrict__ tx1,
                         const float* __restrict__ tx2,
                         const float* __restrict__ hr,
                         const float* __restrict__ tr1,
                         const float* __restrict__ tr2,
                         const float* __restrict__ Wx,
                         const float* __restrict__ Wh,
                         const float* __restrict__ bx,
                         const float* __restrict__ bh,
                         const float* __restrict__ zbuf,
                         const float* __restrict__ h0, float* __restrict__ hout,
                         int n) {
  int wv = (int)(((long long)blockIdx.x * blockDim.x + threadIdx.x) >> 5);
  int lane = threadIdx.x & 31;
  int ntiles = n >> 4;
  if (wv >= ntiles) return;  // wave-uniform
  int row0 = wv << 4;

  v8f acc[4] = {};
  const float* xs[3] = {x, tx1, tx2};
  const float* hs[3] = {hr, tr1, tr2};

#pragma unroll
  for (int kc = 0; kc < 3; ++kc) {
    const float* A = xs[kc];
    const float* W = Wx + (size_t)(2 * 3 + kc) * (CIN * COUT);
    for (int k0 = 0; k0 < CIN; k0 += 4) {
      v2f a = load_a4(A, CIN, row0, k0, lane);
#pragma unroll
      for (int nt = 0; nt < 4; ++nt)
        acc[nt] = wmma4(a, load_b4(W, COUT, k0, nt * 16, lane), acc[nt]);
    }
  }
#pragma unroll
  for (int kc = 0; kc < 3; ++kc) {
    const float* A = hs[kc];
    const float* W = Wh + (size_t)(2 * 3 + kc) * (COUT * COUT);
    for (int k0 = 0; k0 < COUT; k0 += 4) {
      v2f a = load_a4(A, COUT, row0, k0, lane);
#pragma unroll
      for (int nt = 0; nt < 4; ++nt)
        acc[nt] = wmma4(a, load_b4(W, COUT, k0, nt * 16, lane), acc[nt]);
    }
  }

  const int half = lane >> 4, nloc = lane & 15;
#pragma unroll
  for (int nt = 0; nt < 4; ++nt) {
    int col = nt * 16 + nloc;
    float bias = bx[2 * COUT + col] + bh[2 * COUT + col];
#pragma unroll
    for (int i = 0; i < 8; ++i) {
      int row = row0 + i + 8 * half;
      size_t idx = (size_t)row * COUT + col;
      float htil = tanhf(acc[nt][i] + bias);
      float z = zbuf[idx];
      hout[idx] = z * h0[idx] + (1.0f - z) * htil;
    }
  }
}

// y = relu(h) @ Wlin(64x7) + blin ; B padded with zeros to 16 cols
__global__ void lin_y_k(const float* __restrict__ h,
                        const float* __restrict__ Wlin,
                        const float* __restrict__ blin, float* __restrict__ y,
                        int n) {
  int wv = (int)(((long long)blockIdx.x * blockDim.x + threadIdx.x) >> 5);
  int lane = threadIdx.x & 31;
  int ntiles = n >> 4;
  if (wv >= ntiles) return;  // wave-uniform
  int row0 = wv << 4;
  const int half = lane >> 4, nloc = lane & 15;

  v8f acc = {};
  for (int k0 = 0; k0 < COUT; k0 += 4) {
    // relu applied to A on load
    const float* pa = h + (size_t)(row0 + nloc) * COUT + k0 + 2 * half;
    v2f a; a.x = fmaxf(pa[0], 0.0f); a.y = fmaxf(pa[1], 0.0f);
    v2f b;
    if (nloc < 7) {
      const float* pb = Wlin + (size_t)(k0 + 2 * half) * 7 + nloc;
      b.x = pb[0]; b.y = pb[7];
    } else {
      b.x = 0.0f; b.y = 0.0f;
    }
    acc = wmma4(a, b, acc);  // full EXEC: branches reconverge before WMMA
  }

  if (nloc < 7) {
    float bb = blin[nloc];
#pragma unroll
    for (int i = 0; i < 8; ++i) {
      int row = row0 + i + 8 * half;
      y[(size_t)row * 7 + nloc] = acc[i] + bb;
    }
  }
}

// ---------------- host launcher ----------------
extern "C" void kernel_launch(void* const* d_in, const int* in_sizes, int n_in,
                              void* d_out, int out_size, void* d_ws,
                              size_t ws_size, hipStream_t stream) {
  const float* x = (const float*)d_in[0];
  const long long* ei = (const long long*)d_in[1];  // int64 (2,E)
  const float* ew = (const float*)d_in[2];
  const float* h0 = (const float*)d_in[3];
  const float* Wx = (const float*)d_in[4];  // (3,3,16,64)
  const float* Wh = (const float*)d_in[5];  // (3,3,64,64)
  const float* bx = (const float*)d_in[6];  // (3,64)
  const float* bh = (const float*)d_in[7];  // (3,64)
  const float* Wlin = (const float*)d_in[8];  // (64,7)
  const float* blin = (const float*)d_in[9];  // (7,)

  const int n = in_sizes[0] / CIN;  // 50000 (multiple of 16)
  const int E = in_sizes[2];        // 800000

  // workspace carve-out (all offsets float4-aligned)
  float* ws = (float*)d_ws;
  float* deg = ws;                           // n      (becomes dinv)
  float* nw = deg + n;                       // E
  float* tx1 = nw + E;                       // n*16
  float* tx2 = tx1 + (size_t)n * CIN;        // n*16
  float* th1 = tx2 + (size_t)n * CIN;        // n*64 (reused as tr1)
  float* th2 = th1 + (size_t)n * COUT;       // n*64 (reused as tr2)
  float* zbuf = th2 + (size_t)n * COUT;      // n*64
  float* hr = zbuf + (size_t)n * COUT;       // n*64

  float* hout = (float*)d_out;               // n*64
  float* yout = hout + (size_t)n * COUT;     // n*7

  auto nb = [](long long t) { return (int)((t + TB - 1) / TB); };
  const int ntiles = n >> 4;

  // 1) edge normalization
  fill0_k<<<nb(n), TB, 0, stream>>>(deg, n);
  deg_scatter_k<<<nb(E), TB, 0, stream>>>(ew, ei, deg, E);
  dinv_k<<<nb(n), TB, 0, stream>>>(deg, n);
  nw_k<<<nb(E), TB, 0, stream>>>(ew, ei, deg, nw, E);

  // 2) Chebyshev chain on x (C=16): T1 = L~x ; T2 = 2*L~T1 - T0
  fill0_k<<<nb((long long)n * CIN), TB, 0, stream>>>(tx1, (long long)n * CIN);
  prop_scatter_k<<<nb((long long)E * 4), TB, 0, stream>>>(x, nw, ei, tx1, 1.0f,
                                                          E, CIN, 2);
  negcopy_k<<<nb((long long)n * CIN), TB, 0, stream>>>(tx2, x,
                                                       (long long)n * CIN);
  prop_scatter_k<<<nb((long long)E * 4), TB, 0, stream>>>(tx1, nw, ei, tx2,
                                                          2.0f, E, CIN, 2);

  // 3) Chebyshev chain on h0 (C=64)
  fill0_k<<<nb((long long)n * COUT), TB, 0, stream>>>(th1, (long long)n * COUT);
  prop_scatter_k<<<nb((long long)E * 16), TB, 0, stream>>>(h0, nw, ei, th1,
                                                           1.0f, E, COUT, 4);
  negcopy_k<<<nb((long long)n * COUT), TB, 0, stream>>>(th2, h0,
                                                        (long long)n * COUT);
  prop_scatter_k<<<nb((long long)E * 16), TB, 0, stream>>>(th1, nw, ei, th2,
                                                           2.0f, E, COUT, 4);

  // 4) z and r gates (WMMA) -> z, hr = h0*r
  gate_zr_k<<<nb((long long)2 * ntiles * 32), TB, 0, stream>>>(
      x, tx1, tx2, h0, th1, th2, Wx, Wh, bx, bh, zbuf, hr, n);

  // 5) Chebyshev chain on hr (reuse th1/th2 buffers)
  fill0_k<<<nb((long long)n * COUT), TB, 0, stream>>>(th1, (long long)n * COUT);
  prop_scatter_k<<<nb((long long)E * 16), TB, 0, stream>>>(hr, nw, ei, th1,
                                                           1.0f, E, COUT, 4);
  negcopy_k<<<nb((long long)n * COUT), TB, 0, stream>>>(th2, hr,
                                                        (long long)n * COUT);
  prop_scatter_k<<<nb((long long)E * 16), TB, 0, stream>>>(th1, nw, ei, th2,
                                                           2.0f, E, COUT, 4);

  // 6) candidate gate + GRU blend (WMMA) -> h into d_out
  gate_h_k<<<nb((long long)ntiles * 32), TB, 0, stream>>>(
      x, tx1, tx2, hr, th1, th2, Wx, Wh, bx, bh, zbuf, h0, hout, n);

  // 7) y = relu(h) @ Wlin + blin (WMMA) -> y into d_out
  lin_y_k<<<nb((long long)ntiles * 32), TB, 0, stream>>>(hout, Wlin, blin,
                                                         yout, n);
}